// CRF_22428319220244
// MI455X (gfx1250) — compile-verified
//
#include <hip/hip_runtime.h>
#include <hip/hip_bf16.h>

// ---------------------------------------------------------------------------
// CRF Viterbi decode for MI455X (gfx1250, wave32, WMMA).
// B=256, T=2048, F=128, U=48.
// ---------------------------------------------------------------------------

#define CRF_B 256
#define CRF_T 2048
#define CRF_F 128
#define CRF_U 48
#define CRF_BT (CRF_B * CRF_T)          // 524288 rows
#define AS_STRIDE 132                    // 128 + 4 pad: kills LDS bank conflicts

typedef __attribute__((ext_vector_type(2))) float v2f;
typedef __attribute__((ext_vector_type(8))) float v8f;

// ---------------------------------------------------------------------------
// Kernel 1: potentials = x @ W + bias (+left at t==0, +right at t==T-1)
// Tall-skinny GEMM via V_WMMA_F32_16X16X4_F32.
// Block: 256 threads (8 waves). Each wave computes a 16-row x 48-col strip.
// Grid: BT/128 = 4096 blocks.
// ---------------------------------------------------------------------------
__global__ __launch_bounds__(256) void crf_potentials_gemm(
    const float* __restrict__ x, const float* __restrict__ W,
    const float* __restrict__ bias, const float* __restrict__ lb,
    const float* __restrict__ rb, float* __restrict__ pot) {
  // W in K-pair-interleaved layout: Wp[(k>>1)*U + u] = {W[k][u], W[k+1][u]}
  __shared__ __align__(16) float WpF[(CRF_F / 2) * CRF_U * 2];  // 24 KB
  __shared__ __align__(16) float As[8][16 * AS_STRIDE];         // 8 x 8.25 KB

  const int tid = threadIdx.x;
  const int wave = tid >> 5;
  const int lane = tid & 31;
  const int l15 = lane & 15;
  const int half = lane >> 4;

  // Stage W (coalesced global reads, one-time scatter to paired layout).
  for (int i = tid; i < CRF_F * CRF_U; i += 256) {
    const int k = i / CRF_U;
    const int u = i - k * CRF_U;
    WpF[((k >> 1) * CRF_U + u) * 2 + (k & 1)] = W[i];
  }

  // Stage this wave's 16 x 128 A tile: 32 lanes x float4 = one full row/load.
  const int rowBase = blockIdx.x * 128 + wave * 16;
  {
    const float4* xr = reinterpret_cast<const float4*>(x);  // 32 float4 / row
    #pragma unroll
    for (int r = 0; r < 16; ++r) {
      const float4 v = xr[(rowBase + r) * 32 + lane];
      *reinterpret_cast<float4*>(&As[wave][r * AS_STRIDE + lane * 4]) = v;
    }
  }
  __syncthreads();

  v8f acc[3] = {v8f{}, v8f{}, v8f{}};
  const float* aRow = &As[wave][l15 * AS_STRIDE + half * 2];
  const v2f* wp = reinterpret_cast<const v2f*>(WpF);

  #pragma unroll 8
  for (int ks = 0; ks < 32; ++ks) {  // K = 128 = 32 steps of 4
    const v2f a = *reinterpret_cast<const v2f*>(aRow + ks * 4);
    const int kp = ks * 2 + half;
    const v2f b0 = wp[kp * CRF_U + l15];
    const v2f b1 = wp[kp * CRF_U + 16 + l15];
    const v2f b2 = wp[kp * CRF_U + 32 + l15];
    acc[0] = __builtin_amdgcn_wmma_f32_16x16x4_f32(false, a, false, b0,
                                                   (short)0, acc[0], false, false);
    acc[1] = __builtin_amdgcn_wmma_f32_16x16x4_f32(false, a, false, b1,
                                                   (short)0, acc[1], false, false);
    acc[2] = __builtin_amdgcn_wmma_f32_16x16x4_f32(false, a, false, b2,
                                                   (short)0, acc[2], false, false);
  }

  // Epilogue: bias + boundary energies, store to potentials.
  #pragma unroll
  for (int nt = 0; nt < 3; ++nt) {
    const int col = nt * 16 + l15;
    const float bv = bias[col];
    const float lbv = lb[col];
    const float rbv = rb[col];
    #pragma unroll
    for (int r = 0; r < 8; ++r) {
      const int row = rowBase + r + half * 8;  // C layout: VGPR r -> M=r / r+8
      const int t = row & (CRF_T - 1);
      float v = acc[nt][r] + bv;
      if (t == 0) v += lbv;
      if (t == CRF_T - 1) v += rbv;
      pot[row * CRF_U + col] = v;
    }
  }
}

// ---------------------------------------------------------------------------
// Kernel 2: Viterbi forward scan. One 64-thread block per batch; lane u owns
// tag `next=u`. Chain-kernel column lives in 48 VGPRs; alpha ping-pongs in
// LDS (12 x b128 broadcast reads / step). Potentials (fit in 192MB L2) are
// prefetched 2 steps deep to hide L2 latency across the barrier.
// Backpointers -> u8 [b][t][u] in workspace.
// ---------------------------------------------------------------------------
__global__ __launch_bounds__(64) void crf_viterbi_fwd(
    const float* __restrict__ pot, const float* __restrict__ K,
    float* __restrict__ tags, int* __restrict__ lastTag,
    unsigned char* __restrict__ bp) {
  __shared__ __align__(16) float aBuf[2][CRF_U];
  const int b = blockIdx.x;
  const int u = threadIdx.x;
  const bool act = (u < CRF_U);

  float kreg[CRF_U];
  float pv1 = 0.f, pv2 = 0.f;
  if (act) {
    #pragma unroll
    for (int p = 0; p < CRF_U; ++p) kreg[p] = K[p * CRF_U + u];  // coalesced
    aBuf[0][u] = pot[(b * CRF_T + 0) * CRF_U + u];
    pv1 = pot[(b * CRF_T + 1) * CRF_U + u];
    pv2 = pot[(b * CRF_T + 2) * CRF_U + u];
  }
  __syncthreads();

  int par = 0;
  for (int t = 1; t < CRF_T; ++t) {
    if (act) {
      const float pv = pv1;  // 2-deep software pipeline on the L2-resident pot
      pv1 = pv2;
      if (t + 2 < CRF_T) pv2 = pot[(b * CRF_T + (t + 2)) * CRF_U + u];
      float m = -__builtin_inff();
      int bi = 0;
      const float4* ap = reinterpret_cast<const float4*>(aBuf[par]);
      #pragma unroll
      for (int j = 0; j < 12; ++j) {
        const float4 q = ap[j];  // LDS broadcast
        const int p = j * 4;
        float s;
        s = q.x + kreg[p + 0]; if (s > m) { m = s; bi = p + 0; }
        s = q.y + kreg[p + 1]; if (s > m) { m = s; bi = p + 1; }
        s = q.z + kreg[p + 2]; if (s > m) { m = s; bi = p + 2; }
        s = q.w + kreg[p + 3]; if (s > m) { m = s; bi = p + 3; }
      }
      bp[(b * (CRF_T - 1) + (t - 1)) * CRF_U + u] = (unsigned char)bi;
      aBuf[par ^ 1][u] = m + pv;
    }
    __syncthreads();
    par ^= 1;
  }

  if (u == 0) {  // argmax of final alpha -> last tag
    float m = aBuf[par][0];
    int bi = 0;
    for (int p = 1; p < CRF_U; ++p) {
      const float s = aBuf[par][p];
      if (s > m) { m = s; bi = p; }
    }
    lastTag[b] = bi;
    tags[b * CRF_T + (CRF_T - 1)] = (float)bi;
  }
}

// ---------------------------------------------------------------------------
// Kernel 3: traceback. 8 blocks x 256 threads; each block owns 32 batches.
// bp rows depend only on t (not on the running tag), so all 256 threads
// bulk-stage 48-timestep chunks into a double-buffered LDS ring while the
// 32 lanes of wave 0 run the serial byte-select chain out of LDS. One
// barrier per chunk; staging of chunk c+1 overlaps selects of chunk c.
// Also fills seq_len and copies chain_kernel through (block 0).
// ---------------------------------------------------------------------------
#define TB_CH 48                          // timesteps per chunk
#define TB_NCH ((CRF_T - 1 + TB_CH - 1) / TB_CH)           // 43
#define TB_STRIDE_W (TB_CH * 12 + 4)      // words per batch region (pad 16B)
#define TB_K ((TB_CH * 3) / 8)            // uint4 loads per staging thread: 18

__global__ __launch_bounds__(256) void crf_viterbi_bwd(
    const unsigned char* __restrict__ bp, const int* __restrict__ lastTag,
    const float* __restrict__ K, float* __restrict__ tags,
    float* __restrict__ seqlen, float* __restrict__ chainOut) {
  __shared__ __align__(16) unsigned int ring[2][32 * TB_STRIDE_W];  // ~145 KB
  const int tid = threadIdx.x;
  const int g = tid >> 3;   // staging: batch slot 0..31 (8 threads each)
  const int sub = tid & 7;
  const int gb = blockIdx.x * 32 + g;
  const uint4* gsrc =
      reinterpret_cast<const uint4*>(bp) + (size_t)gb * ((CRF_T - 1) * 3);

  // ---- stage chunk c into ring[p]: two phases so loads batch one wait ----
  auto stage = [&](int c, int p) {
    const int tHi = (CRF_T - 2) - c * TB_CH;
    const int tLo = tHi - (TB_CH - 1);  // may go negative in last chunk
    uint4 v[TB_K];
    #pragma unroll
    for (int k = 0; k < TB_K; ++k) {
      const int m = sub + 8 * k;        // 0..143, contiguous across 8 threads
      const int t = tLo + m / 3;
      v[k] = (t >= 0) ? gsrc[(size_t)t * 3 + (m % 3)]
                      : uint4{0u, 0u, 0u, 0u};
    }
    uint4* dst = reinterpret_cast<uint4*>(&ring[p][g * TB_STRIDE_W]);
    #pragma unroll
    for (int k = 0; k < TB_K; ++k) dst[sub + 8 * k] = v[k];
  };

  if (blockIdx.x == 0) {  // side outputs (deterministic, done once per call)
    seqlen[tid] = (float)CRF_T;
    for (int i = tid; i < CRF_U * CRF_U; i += 256) chainOut[i] = K[i];
  }

  int tag = 0;
  if (tid < 32) tag = lastTag[blockIdx.x * 32 + tid];

  stage(0, 0);
  __syncthreads();
  for (int c = 0; c < TB_NCH; ++c) {
    if (c + 1 < TB_NCH) stage(c + 1, (c + 1) & 1);
    if (tid < 32) {  // wave 0: serial select chain out of LDS
      const int own = blockIdx.x * 32 + tid;
      const int tHi = (CRF_T - 2) - c * TB_CH;
      const int tLo = tHi - (TB_CH - 1);
      const int tMin = (tLo < 0) ? 0 : tLo;
      const unsigned int* base = &ring[c & 1][tid * TB_STRIDE_W];
      for (int t = tHi; t >= tMin; --t) {
        const int jj = t - tLo;
        const uint4 r0 = *reinterpret_cast<const uint4*>(base + jj * 12 + 0);
        const uint4 r1 = *reinterpret_cast<const uint4*>(base + jj * 12 + 4);
        const uint4 r2 = *reinterpret_cast<const uint4*>(base + jj * 12 + 8);
        const int wi = tag >> 2;
        const unsigned w =
            wi == 0 ? r0.x : wi == 1 ? r0.y : wi == 2 ? r0.z : wi == 3 ? r0.w :
            wi == 4 ? r1.x : wi == 5 ? r1.y : wi == 6 ? r1.z : wi == 7 ? r1.w :
            wi == 8 ? r2.x : wi == 9 ? r2.y : wi == 10 ? r2.z : r2.w;
        tag = (int)((w >> ((tag & 3) * 8)) & 0xFFu);
        tags[own * CRF_T + t] = (float)tag;
      }
    }
    __syncthreads();
  }
}

// ---------------------------------------------------------------------------
extern "C" void kernel_launch(void* const* d_in, const int* in_sizes, int n_in,
                              void* d_out, int out_size, void* d_ws,
                              size_t ws_size, hipStream_t stream) {
  const float* x  = (const float*)d_in[0];
  const float* W  = (const float*)d_in[1];
  const float* bv = (const float*)d_in[2];
  const float* K  = (const float*)d_in[3];
  const float* lb = (const float*)d_in[4];
  const float* rb = (const float*)d_in[5];

  float* out  = (float*)d_out;
  float* tags = out;                           // [B*T]
  float* pot  = out + CRF_BT;                  // [B*T*U]
  float* seql = pot + (size_t)CRF_BT * CRF_U;  // [B]
  float* chn  = seql + CRF_B;                  // [U*U]

  int* lastTag = (int*)d_ws;                        // 256 ints
  unsigned char* bp = (unsigned char*)d_ws + 1024;  // ~25.2 MB, 16B aligned

  crf_potentials_gemm<<<CRF_BT / 128, 256, 0, stream>>>(x, W, bv, lb, rb, pot);
  crf_viterbi_fwd<<<CRF_B, 64, 0, stream>>>(pot, K, tags, lastTag, bp);
  crf_viterbi_bwd<<<CRF_B / 32, 256, 0, stream>>>(bp, lastTag, K, tags, seql, chn);
}